// GAEDecoder_36051955482714
// MI455X (gfx1250) — compile-verified
//
#include <hip/hip_runtime.h>
#include <hip/hip_bf16.h>

// ---------------------------------------------------------------------------
// 2-layer GCN for MI455X (gfx1250, wave32).
//   layer: h = x @ W ; out[d] += inv[s]*inv[d]*h[s] (+ self loop) ; +bias (; relu)
// GEMMs use V_WMMA_F32_16X16X4_F32 (full f32 precision -- GEMM is ~1.6 GFLOP
// total, far from the bottleneck; aggregation is the memory-bound part).
// Aggregation uses float4 gathers + global_atomic_add_f32 scatters; working
// set (h0 25.6MB, agg 25.6MB) fits in the 192MB L2.
// ---------------------------------------------------------------------------

typedef float v2f __attribute__((ext_vector_type(2)));
typedef float v8f __attribute__((ext_vector_type(8)));

#define N_IN 64
#define N_HID 128
#define N_OUT 64

// ---------------- degree / norm kernels ----------------

__global__ void k_deg_init(float* __restrict__ deg, int n) {
    int i = blockIdx.x * blockDim.x + threadIdx.x;
    if (i < n) deg[i] = 1.0f;               // self loop
}

__global__ void k_deg_count(const int* __restrict__ dst, float* __restrict__ deg, int E) {
    int e = blockIdx.x * blockDim.x + threadIdx.x;
    if (e < E) atomicAdd(&deg[dst[e]], 1.0f);
}

__global__ void k_deg_rsqrt(float* __restrict__ deg, int n) {
    int i = blockIdx.x * blockDim.x + threadIdx.x;
    if (i < n) deg[i] = rsqrtf(fmaxf(deg[i], 1.0f));   // deg -> D^-1/2
}

// ---------------- f32 WMMA GEMM: C[M,N] = A[M,K] @ B[K,N] ----------------
// One wave per 16x16 output tile. M,N,K must be multiples of 16/16/4
// (true here: M=50000, K in {64,128}, N in {128,64}).
//
// Fragment layouts per CDNA5 ISA 7.12.2 (wave32):
//   A 16x4 f32 : lane<16 -> row=lane,   v.{x,y} = K {k,k+1}
//                lane>=16 -> row=lane-16, v.{x,y} = K {k+2,k+3}
//   B 4x16 f32 : lane<16 -> col=lane,   v.{x,y} = K {k,k+1}
//                lane>=16 -> col=lane-16, v.{x,y} = K {k+2,k+3}
//   C/D        : vgpr j -> row = 8*half + j, col = lane&15
template <int K, int N>
__global__ void k_gemm_wmma_f32(const float* __restrict__ A,
                                const float* __restrict__ B,
                                float* __restrict__ C, int M) {
    const int lane = threadIdx.x & 31;
    const int wave = threadIdx.x >> 5;
    const int wavesPerBlock = blockDim.x >> 5;
    const int tile = blockIdx.x * wavesPerBlock + wave;

    const int nTiles = N / 16;
    const int mTiles = M / 16;
    if (tile >= mTiles * nTiles) return;    // wave-uniform: EXEC stays all-1s

    const int row0 = (tile / nTiles) * 16;
    const int col0 = (tile % nTiles) * 16;
    const int half = lane >> 4;             // 0: lanes 0-15, 1: lanes 16-31
    const int l16  = lane & 15;

    v8f acc = {0.f, 0.f, 0.f, 0.f, 0.f, 0.f, 0.f, 0.f};

#pragma unroll
    for (int k = 0; k < K; k += 4) {
        const float* ap = A + (size_t)(row0 + l16) * K + (k + 2 * half);
        v2f a;
        a.x = ap[0];
        a.y = ap[1];
        const float* bp = B + (size_t)(k + 2 * half) * N + (col0 + l16);
        v2f b;
        b.x = bp[0];
        b.y = bp[N];
        acc = __builtin_amdgcn_wmma_f32_16x16x4_f32(
            /*neg_a=*/false, a, /*neg_b=*/false, b,
            /*c_mod=*/(short)0, acc, /*reuse_a=*/false, /*reuse_b=*/false);
    }

#pragma unroll
    for (int j = 0; j < 8; ++j) {
        C[(size_t)(row0 + 8 * half + j) * N + (col0 + l16)] = acc[j];
    }
}

// ---------------- aggregation kernels ----------------

// out[i] = inv[i]^2 * h[i]  (+ bias) : self-loop contribution, deterministic init
template <int C>
__global__ void k_agg_init(const float* __restrict__ h, const float* __restrict__ inv,
                           const float* __restrict__ bias, float* __restrict__ out, int n) {
    const int C4 = C / 4;
    int idx = blockIdx.x * blockDim.x + threadIdx.x;
    if (idx >= n * C4) return;
    int i  = idx / C4;
    int c4 = idx % C4;
    float s = inv[i];
    s = s * s;
    float4 v = ((const float4*)(h + (size_t)i * C))[c4];
    float4 r;
    r.x = s * v.x; r.y = s * v.y; r.z = s * v.z; r.w = s * v.w;
    if (bias != nullptr) {
        float4 bv = ((const float4*)bias)[c4];
        r.x += bv.x; r.y += bv.y; r.z += bv.z; r.w += bv.w;
    }
    ((float4*)(out + (size_t)i * C))[c4] = r;
}

// out[dst] += inv[src]*inv[dst] * h[src] : one float4 per thread
template <int C>
__global__ void k_agg_edges(const float* __restrict__ h, const int* __restrict__ src,
                            const int* __restrict__ dst, const float* __restrict__ inv,
                            float* __restrict__ out, int E) {
    const int C4 = C / 4;
    int idx = blockIdx.x * blockDim.x + threadIdx.x;
    if (idx >= E * C4) return;
    int e  = idx / C4;
    int c4 = idx % C4;
    int s = src[e];
    int d = dst[e];
    float nrm = inv[s] * inv[d];
    float4 v = ((const float4*)(h + (size_t)s * C))[c4];
    float* op = out + (size_t)d * C + 4 * c4;
    atomicAdd(op + 0, nrm * v.x);
    atomicAdd(op + 1, nrm * v.y);
    atomicAdd(op + 2, nrm * v.z);
    atomicAdd(op + 3, nrm * v.w);
}

// a = relu(a + bias), in place
template <int C>
__global__ void k_bias_relu(float* __restrict__ a, const float* __restrict__ bias, int n) {
    int idx = blockIdx.x * blockDim.x + threadIdx.x;
    if (idx >= n * C) return;
    int c = idx % C;
    a[idx] = fmaxf(a[idx] + bias[c], 0.0f);
}

// ---------------- launch ----------------

extern "C" void kernel_launch(void* const* d_in, const int* in_sizes, int n_in,
                              void* d_out, int out_size, void* d_ws, size_t ws_size,
                              hipStream_t stream) {
    const float* x    = (const float*)d_in[0];
    const int*   eidx = (const int*)d_in[1];
    const float* W1   = (const float*)d_in[2];
    const float* b1   = (const float*)d_in[3];
    const float* W2   = (const float*)d_in[4];
    const float* b2   = (const float*)d_in[5];
    float*       out  = (float*)d_out;

    const int n = in_sizes[0] / N_IN;       // 50000 (multiple of 16)
    const int E = in_sizes[1] / 2;          // 800000
    const int* src = eidx;
    const int* dst = eidx + E;

    // workspace layout (all 512B aligned)
    char* ws = (char*)d_ws;
    const size_t OFF_INV = 0;                                  // n f32
    const size_t OFF_H0  = (((size_t)n * 4) + 511) & ~(size_t)511;   // n*128 f32
    const size_t OFF_AGG = OFF_H0 + (size_t)n * N_HID * 4;           // n*128 f32
    float* inv  = (float*)(ws + OFF_INV);
    float* h0   = (float*)(ws + OFF_H0);    // layer-1 linear out; later reused for h2
    float* agg1 = (float*)(ws + OFF_AGG);   // layer-1 aggregate -> relu'd hidden

    const int TPB = 256;

    // degrees -> D^-1/2
    k_deg_init<<<(n + TPB - 1) / TPB, TPB, 0, stream>>>(inv, n);
    k_deg_count<<<(E + TPB - 1) / TPB, TPB, 0, stream>>>(dst, inv, E);
    k_deg_rsqrt<<<(n + TPB - 1) / TPB, TPB, 0, stream>>>(inv, n);

    // layer 1: h0 = x @ W1   (M=n, K=64, N=128) -> 16x16 tiles, 4 waves/block
    {
        int tiles = (n / 16) * (N_HID / 16);
        int blocks = (tiles + 3) / 4;
        k_gemm_wmma_f32<N_IN, N_HID><<<blocks, 128, 0, stream>>>(x, W1, h0, n);
    }
    // aggregate + bias + relu
    k_agg_init<N_HID><<<((n * (N_HID / 4)) + TPB - 1) / TPB, TPB, 0, stream>>>(
        h0, inv, nullptr, agg1, n);
    k_agg_edges<N_HID><<<((E * (N_HID / 4)) + TPB - 1) / TPB, TPB, 0, stream>>>(
        h0, src, dst, inv, agg1, E);
    k_bias_relu<N_HID><<<((n * N_HID) + TPB - 1) / TPB, TPB, 0, stream>>>(agg1, b1, n);

    // layer 2: h2 = h1 @ W2  (M=n, K=128, N=64), h2 reuses the h0 slot
    float* h2 = h0;
    {
        int tiles = (n / 16) * (N_OUT / 16);
        int blocks = (tiles + 3) / 4;
        k_gemm_wmma_f32<N_HID, N_OUT><<<blocks, 128, 0, stream>>>(agg1, W2, h2, n);
    }
    // aggregate + bias straight into d_out
    k_agg_init<N_OUT><<<((n * (N_OUT / 4)) + TPB - 1) / TPB, TPB, 0, stream>>>(
        h2, inv, b2, out, n);
    k_agg_edges<N_OUT><<<((E * (N_OUT / 4)) + TPB - 1) / TPB, TPB, 0, stream>>>(
        h2, src, dst, inv, out, E);

    (void)n_in; (void)out_size; (void)ws_size;
}